// DecoderRnn_68221260529926
// MI455X (gfx1250) — compile-verified
//
#include <hip/hip_runtime.h>
#include <math.h>

// ---------------------------------------------------------------------------
// GRU decoder step for MI455X (gfx1250, wave32).
// Bandwidth-bound GEMV workload (~256MB weights / call, ~11us floor @23.3TB/s).
// Matvecs are routed through V_WMMA_F32_16X16X4_F32: one wave computes 16 dot
// products, each lane issues one coalesced global_load_b64 + one ds_load_b64
// per WMMA. B is the input vector replicated across all 16 columns, so every
// column of D holds the same 16 results and each half-wave ends with all of
// its 8 results resident in the v8f accumulator (no cross-lane extraction).
// ---------------------------------------------------------------------------

typedef float v2f __attribute__((ext_vector_type(2)));
typedef float v8f __attribute__((ext_vector_type(8)));

#define HDIM 1024
#define VOCAB 50257

// 16-row GEMV tile: returns, for this lane, results for rows
// row0 + (lane>=16 ? 8 : 0) + m  (m = 0..7) in acc[m].
// W row `r` load clamped to rowMax to keep EXEC all-ones on the tail tile.
__device__ __forceinline__ v8f matvec16_wmma(const float* __restrict__ W,
                                             int row0,
                                             const float* __restrict__ xs,
                                             int lane, int rowMax) {
  const int half = lane >> 4;        // 0: K=0,1 ; 1: K=2,3  (A and B layouts)
  const int mrow = lane & 15;
  int r = row0 + mrow;
  if (r > rowMax) r = rowMax;        // duplicate loads on tail, discarded later
  const float* wr = W + (size_t)r * HDIM + 2 * half;
  const float* xp = xs + 2 * half;
  v8f acc = {0.f, 0.f, 0.f, 0.f, 0.f, 0.f, 0.f, 0.f};
#pragma unroll 4
  for (int k = 0; k < HDIM; k += 4) {
    v2f a = *(const v2f*)(wr + k);   // global_load_b64, coalesced over 16 rows
    v2f b = *(const v2f*)(xp + k);   // ds_load_b64 broadcast
    acc = __builtin_amdgcn_wmma_f32_16x16x4_f32(
        false, a, false, b, (short)0, acc, false, false);
  }
  return acc;
}

__device__ __forceinline__ float sigmoidf_(float x) {
  return 1.0f / (1.0f + __expf(-x));
}

// ---------------------------------------------------------------------------
__global__ void __launch_bounds__(256) k_embed_relu(const int* __restrict__ tok,
                                                    const float* __restrict__ emb,
                                                    float* __restrict__ x) {
  const int t = tok[0];
  const float* e = emb + (size_t)t * HDIM;
  for (int i = threadIdx.x; i < HDIM; i += 256) x[i] = fmaxf(e[i], 0.0f);
}

// One GRU layer. 8 blocks x 8 waves; each wave owns 16 output channels and
// runs 6 WMMA GEMV tiles (r/z/n rows of w_ih and w_hh), then combines gates
// entirely within the wave (results are replicated across each half-wave).
__global__ void __launch_bounds__(256) k_gru_layer(
    const float* __restrict__ x, const float* __restrict__ hprev,
    const float* __restrict__ wih, const float* __restrict__ whh,
    const float* __restrict__ bih, const float* __restrict__ bhh,
    float* __restrict__ xnext, float* __restrict__ hout) {
  __shared__ float xs[HDIM];
  __shared__ float hs[HDIM];
  for (int i = threadIdx.x; i < HDIM; i += 256) {
    xs[i] = x[i];
    hs[i] = hprev[i];
  }
  __syncthreads();

  const int lane = threadIdx.x & 31;
  const int wave = threadIdx.x >> 5;
  const int j0 = (blockIdx.x * 8 + wave) * 16;   // 0..1008, always in range

  v8f gir = matvec16_wmma(wih, 0 * HDIM + j0, xs, lane, 3 * HDIM - 1);
  v8f giz = matvec16_wmma(wih, 1 * HDIM + j0, xs, lane, 3 * HDIM - 1);
  v8f gin = matvec16_wmma(wih, 2 * HDIM + j0, xs, lane, 3 * HDIM - 1);
  v8f ghr = matvec16_wmma(whh, 0 * HDIM + j0, hs, lane, 3 * HDIM - 1);
  v8f ghz = matvec16_wmma(whh, 1 * HDIM + j0, hs, lane, 3 * HDIM - 1);
  v8f ghn = matvec16_wmma(whh, 2 * HDIM + j0, hs, lane, 3 * HDIM - 1);

  const int half = lane >> 4;
  const int jb = j0 + half * 8;
  float res[8];
#pragma unroll
  for (int m = 0; m < 8; ++m) {
    const int j = jb + m;
    const float rg = gir[m] + bih[j]            + ghr[m] + bhh[j];
    const float zg = giz[m] + bih[HDIM + j]     + ghz[m] + bhh[HDIM + j];
    const float ng = gin[m] + bih[2 * HDIM + j];
    const float nh = ghn[m] + bhh[2 * HDIM + j];
    const float rr = sigmoidf_(rg);
    const float zz = sigmoidf_(zg);
    const float nn = tanhf(ng + rr * nh);
    res[m] = (1.0f - zz) * nn + zz * hs[j];
  }
  if ((lane & 15) == 0) {            // lane 0 -> rows jb..jb+7, lane 16 -> +8
#pragma unroll
    for (int m = 0; m < 8; ++m) {
      xnext[jb + m] = res[m];
      hout[jb + m]  = res[m];
    }
  }
}

// Output projection: logits[r] = w_out[r,:] . x + b_out[r]. One 16-row WMMA
// tile per wave; 3142 tiles over 393 blocks saturate HBM with b64 loads.
__global__ void __launch_bounds__(256) k_logits(const float* __restrict__ x,
                                                const float* __restrict__ wout,
                                                const float* __restrict__ bout,
                                                float* __restrict__ logits) {
  __shared__ float xs[HDIM];
  for (int i = threadIdx.x; i < HDIM; i += 256) xs[i] = x[i];
  __syncthreads();

  const int lane = threadIdx.x & 31;
  const int wave = threadIdx.x >> 5;
  const int row0 = (blockIdx.x * 8 + wave) * 16;
  if (row0 >= VOCAB) return;         // whole-wave exit, EXEC stays uniform

  v8f acc = matvec16_wmma(wout, row0, xs, lane, VOCAB - 1);

  const int half = lane >> 4;
  const int rb = row0 + half * 8;
  if ((lane & 15) == 0) {
#pragma unroll
    for (int m = 0; m < 8; ++m) {
      const int r = rb + m;
      if (r < VOCAB) logits[r] = acc[m] + bout[r];
    }
  }
}

// In-place log_softmax over logits[0..V). Single block of 1024 threads;
// logits (200KB) sit in L2 between passes.
__global__ void __launch_bounds__(1024) k_logsoftmax(float* __restrict__ logits) {
  __shared__ float sred[32];
  __shared__ float sscalar;
  const int tid = threadIdx.x;
  const int lane = tid & 31;
  const int wid = tid >> 5;

  float lmax = -INFINITY;
  for (int i = tid; i < VOCAB; i += 1024) lmax = fmaxf(lmax, logits[i]);
#pragma unroll
  for (int off = 16; off > 0; off >>= 1) lmax = fmaxf(lmax, __shfl_xor(lmax, off));
  if (lane == 0) sred[wid] = lmax;
  __syncthreads();
  if (tid == 0) {
    float m = sred[0];
    for (int i = 1; i < 32; ++i) m = fmaxf(m, sred[i]);
    sscalar = m;
  }
  __syncthreads();
  const float gmax = sscalar;

  float lsum = 0.0f;
  for (int i = tid; i < VOCAB; i += 1024) lsum += __expf(logits[i] - gmax);
#pragma unroll
  for (int off = 16; off > 0; off >>= 1) lsum += __shfl_xor(lsum, off);
  __syncthreads();
  if (lane == 0) sred[wid] = lsum;
  __syncthreads();
  if (tid == 0) {
    float s = 0.0f;
    for (int i = 0; i < 32; ++i) s += sred[i];
    sscalar = gmax + logf(s);
  }
  __syncthreads();
  const float lse = sscalar;

  for (int i = tid; i < VOCAB; i += 1024) logits[i] = logits[i] - lse;
}

// ---------------------------------------------------------------------------
extern "C" void kernel_launch(void* const* d_in, const int* in_sizes, int n_in,
                              void* d_out, int out_size, void* d_ws, size_t ws_size,
                              hipStream_t stream) {
  (void)in_sizes; (void)n_in; (void)out_size; (void)ws_size;
  const int*   tok    = (const int*)d_in[0];     // token id (low word valid)
  const float* hidden = (const float*)d_in[1];   // [L,1,H]
  const float* emb    = (const float*)d_in[2];   // [V,H]
  const float* w_ih   = (const float*)d_in[3];   // [L,3H,H]
  const float* w_hh   = (const float*)d_in[4];   // [L,3H,H]
  const float* b_ih   = (const float*)d_in[5];   // [L,3H]
  const float* b_hh   = (const float*)d_in[6];   // [L,3H]
  const float* w_out  = (const float*)d_in[7];   // [V,H]
  const float* b_out  = (const float*)d_in[8];   // [V]

  float* out = (float*)d_out;                    // [V logprobs][2H hidden]
  float* ws  = (float*)d_ws;
  float* x0 = ws;                                // embedded input
  float* h0 = ws + HDIM;                         // layer-0 hidden out
  float* h1 = ws + 2 * HDIM;                     // layer-1 hidden out

  k_embed_relu<<<1, 256, 0, stream>>>(tok, emb, x0);

  k_gru_layer<<<8, 256, 0, stream>>>(x0, hidden, w_ih, w_hh, b_ih, b_hh,
                                     h0, out + VOCAB);
  k_gru_layer<<<8, 256, 0, stream>>>(h0, hidden + HDIM,
                                     w_ih + 3 * HDIM * HDIM,
                                     w_hh + 3 * HDIM * HDIM,
                                     b_ih + 3 * HDIM, b_hh + 3 * HDIM,
                                     h1, out + VOCAB + HDIM);

  const int tiles  = (VOCAB + 15) / 16;          // 3142
  const int blocks = (tiles + 7) / 8;            // 393
  k_logits<<<blocks, 256, 0, stream>>>(h1, w_out, b_out, out);

  k_logsoftmax<<<1, 1024, 0, stream>>>(out);
}